// Y_DECODER_58506044506607
// MI455X (gfx1250) — compile-verified
//
#include <hip/hip_runtime.h>
#include <hip/hip_bf16.h>

typedef float v2f __attribute__((ext_vector_type(2)));
typedef float v8f __attribute__((ext_vector_type(8)));

#define NNODES 50000
#define NEDGES 1600000
#define FIN    256
#define HID    128
#define NCLS   40
#define NPAD   48

__device__ __forceinline__ float waveReduceSum(float v) {
#pragma unroll
  for (int m = 16; m >= 1; m >>= 1) v += __shfl_xor(v, m, 32);
  return v;
}

// ---- per-row inverse L2 norm (one wave per row) -----------------------------
__global__ void k_rownorm(const float* __restrict__ X, float* __restrict__ invn,
                          int n, int cols) {
  int w = blockIdx.x * (blockDim.x >> 5) + (threadIdx.x >> 5);
  if (w >= n) return;
  int lane = threadIdx.x & 31;
  const float* row = X + (size_t)w * cols;
  float s = 0.f;
  for (int j = lane; j < cols; j += 32) { float v = row[j]; s += v * v; }
  s = waveReduceSum(s);
  if (lane == 0) {
    float nrm = fmaxf(sqrtf(s), 1e-12f);
    invn[w] = 1.0f / nrm;
  }
}

__global__ void k_zero(float* __restrict__ p, size_t n4) {  // n4 = count of float4
  size_t i = (size_t)blockIdx.x * blockDim.x + threadIdx.x;
  size_t stride = (size_t)gridDim.x * blockDim.x;
  float4 z = make_float4(0.f, 0.f, 0.f, 0.f);
  float4* p4 = (float4*)p;
  for (; i < n4; i += stride) p4[i] = z;
}

// ---- f32 WMMA GEMM: C[M x N] = op(A)[M x K] @ B[K x N],  N = NT*16 ----------
// MODE bits (compile-time): 1 = per-row scale (applied to accumulators at store:
//                               diag(s)(A@B) == (diag(s)A)@B, zero K-loop cost)
//                           2 = per-K bias + ReLU on A (fused relu(agg+b1))
//                           4 = per-N bias on C
//                           8 = accumulate into C
// B staged in LDS per K-chunk, pair-swizzled so a lane's fragment
// (B[ka][n], B[ka+1][n]) is one aligned ds_load_b64 with imm offset.
template <int NT, int K, int MODE>
__global__ void k_gemm_wmma(const float* __restrict__ A, const float* __restrict__ B,
                            float* __restrict__ C, int M,
                            const float* __restrict__ rowScale,
                            const float* __restrict__ abias,
                            const float* __restrict__ bias) {
  constexpr bool USE_SCALE = (MODE & 1) != 0;
  constexpr bool USE_ARELU = (MODE & 2) != 0;
  constexpr bool USE_BIAS  = (MODE & 4) != 0;
  constexpr bool ACCUM     = (MODE & 8) != 0;
  constexpr int N = NT * 16;
  constexpr int KCHUNK = (K % 32 == 0) ? 32 : K;     // 32, or 40 for the emb GEMM
  constexpr int NCHUNKS = K / KCHUNK;
  constexpr int TSTEPS = KCHUNK / 4;                 // 8 or 10
  constexpr int GROUP = (TSTEPS % 8 == 0) ? 8 : TSTEPS / 2;  // 8, or 5 (K=40)
  constexpr int NV4 = (KCHUNK * N) / 4;              // float4 elems to stage
  constexpr int STAGE4 = (NV4 + 255) / 256;          // per-thread float4 iters
  __shared__ float ldsB[KCHUNK * N];

  const int lane = threadIdx.x & 31;
  const int wid  = threadIdx.x >> 5;
  const int tm   = blockIdx.x * (blockDim.x >> 5) + wid;  // 16-row tile index
  const bool active = tm < (M >> 4);                      // wave-uniform
  const int h = lane >> 4;                                // half-wave select
  const int l = lane & 15;
  const int row = (tm << 4) + l;

  v8f acc[NT];
  if (active) {
#pragma unroll
    for (int j = 0; j < NT; ++j)
#pragma unroll
      for (int v = 0; v < 8; ++v)
        acc[j][v] = ACCUM ? C[(size_t)((tm << 4) + v + 8 * h) * N + j * 16 + l] : 0.f;
  }
  const float* Arow = A + (size_t)row * K;

  for (int c = 0; c < NCHUNKS; ++c) {
    const int k0 = c * KCHUNK;
    __syncthreads();
#pragma unroll
    for (int i = 0; i < STAGE4; ++i) {
      int idx4 = i * 256 + threadIdx.x;
      if (NV4 % 256 == 0 || idx4 < NV4) {
        int f = idx4 * 4;
        int kk = f / N, n = f - kk * N;               // 4 floats share kk (4|N)
        float4 q = *(const float4*)(B + (size_t)(k0 + kk) * N + n);
        float* d = ldsB + ((kk >> 1) * N + n) * 2 + (kk & 1);
        d[0] = q.x; d[2] = q.y; d[4] = q.z; d[6] = q.w;
      }
    }
    __syncthreads();
    if (active) {
#pragma unroll
      for (int g = 0; g < TSTEPS; g += GROUP) {
        v2f areg[GROUP];                 // short-lived A slice (<=16 VGPRs)
#pragma unroll
        for (int u = 0; u < GROUP; ++u) {
          const int ka = k0 + (g + u) * 4 + 2 * h;
          v2f a;
          a.x = Arow[ka];
          a.y = Arow[ka + 1];
          if (USE_ARELU) {
            a.x = fmaxf(a.x + abias[ka], 0.f);
            a.y = fmaxf(a.y + abias[ka + 1], 0.f);
          }
          areg[u] = a;
        }
#pragma unroll
        for (int u = 0; u < GROUP; ++u) {
          const int ka = (g + u) * 4 + 2 * h;
          const float* bp = ldsB + (ka >> 1) * (2 * N) + 2 * l;
#pragma unroll
          for (int j = 0; j < NT; ++j) {
            v2f b = *(const v2f*)(bp + 32 * j);       // ds_load_b64, imm offset
            acc[j] = __builtin_amdgcn_wmma_f32_16x16x4_f32(
                false, areg[u], false, b, (short)0, acc[j], false, false);
          }
        }
      }
    }
  }

  if (active) {
    float sc[8];
#pragma unroll
    for (int v = 0; v < 8; ++v)
      sc[v] = USE_SCALE ? rowScale[(tm << 4) + v + 8 * h] : 1.0f;
#pragma unroll
    for (int j = 0; j < NT; ++j)
#pragma unroll
      for (int v = 0; v < 8; ++v) {
        const int cn = j * 16 + l;
        float o = acc[j][v];
        if (USE_SCALE) o *= sc[v];
        if (USE_BIAS) o += bias[cn];
        C[(size_t)((tm << 4) + v + 8 * h) * N + cn] = o;  // D row = v + 8*h
      }
  }
}

// ---- SPMM: agg[dst] += w * X[src], one wave per edge, float4 per lane -------
__global__ void k_spmm(const int* __restrict__ src, const int* __restrict__ dst,
                       const float* __restrict__ ew, const float* __restrict__ X,
                       float* __restrict__ agg, int nE) {
  int e = blockIdx.x * (blockDim.x >> 5) + (threadIdx.x >> 5);
  if (e >= nE) return;
  int lane = threadIdx.x & 31;
  int s = src[e], d = dst[e];
  float w = ew[e];
  const float4* xr = reinterpret_cast<const float4*>(X + (size_t)s * HID);
  float4 v = xr[lane];
  float* out = agg + (size_t)d * HID + lane * 4;
  unsafeAtomicAdd(out + 0, v.x * w);
  unsafeAtomicAdd(out + 1, v.y * w);
  unsafeAtomicAdd(out + 2, v.z * w);
  unsafeAtomicAdd(out + 3, v.w * w);
}

// ---- build padded Wf halves + effective bias --------------------------------
// logits = Yemb@Wf_top + agg2@Wf_bot + (bf + be.Wf_top + b2.Wf_bot)
__global__ void k_prep_head(const float* __restrict__ Wf, const float* __restrict__ bf,
                            const float* __restrict__ be, const float* __restrict__ b2,
                            float* __restrict__ Wtp, float* __restrict__ Wbp,
                            float* __restrict__ bfp) {
  int c = threadIdx.x;
  if (c >= NPAD) return;
  float s = 0.f;
  if (c < NCLS) {
    s = bf[c];
    for (int k = 0; k < HID; ++k) {
      s += be[k] * Wf[(size_t)k * NCLS + c];
      s += b2[k] * Wf[(size_t)(HID + k) * NCLS + c];
    }
  }
  bfp[c] = s;
  for (int k = 0; k < HID; ++k) {
    Wtp[k * NPAD + c] = (c < NCLS) ? Wf[(size_t)k * NCLS + c] : 0.f;
    Wbp[k * NPAD + c] = (c < NCLS) ? Wf[(size_t)(HID + k) * NCLS + c] : 0.f;
  }
}

// ---- softmax over 40 classes, one wave per row ------------------------------
__global__ void k_softmax(const float* __restrict__ logits, float* __restrict__ out,
                          int n) {
  int w = blockIdx.x * (blockDim.x >> 5) + (threadIdx.x >> 5);
  if (w >= n) return;
  int lane = threadIdx.x & 31;
  const float* lr = logits + (size_t)w * NPAD;
  float v0 = lr[lane];
  float v1 = (lane < NCLS - 32) ? lr[32 + lane] : -3.4e38f;
  float m = fmaxf(v0, v1);
#pragma unroll
  for (int x = 16; x >= 1; x >>= 1) m = fmaxf(m, __shfl_xor(m, x, 32));
  float e0 = __expf(v0 - m);
  float e1 = (lane < NCLS - 32) ? __expf(v1 - m) : 0.f;
  float s = waveReduceSum(e0 + e1);
  float inv = 1.0f / s;
  float* orow = out + (size_t)w * NCLS;
  orow[lane] = e0 * inv;
  if (lane < NCLS - 32) orow[32 + lane] = e1 * inv;
}

extern "C" void kernel_launch(void* const* d_in, const int* in_sizes, int n_in,
                              void* d_out, int out_size, void* d_ws, size_t ws_size,
                              hipStream_t stream) {
  const float* features = (const float*)d_in[0];
  const int*   src      = (const int*)d_in[1];
  const int*   dst      = (const int*)d_in[2];
  const float* ew       = (const float*)d_in[3];
  const float* y        = (const float*)d_in[4];
  const float* W1       = (const float*)d_in[5];
  const float* b1       = (const float*)d_in[6];
  const float* W2       = (const float*)d_in[7];
  const float* b2       = (const float*)d_in[8];
  const float* We       = (const float*)d_in[9];
  const float* be       = (const float*)d_in[10];
  const float* Wf       = (const float*)d_in[11];
  const float* bf       = (const float*)d_in[12];
  float* out = (float*)d_out;
  (void)in_sizes; (void)n_in; (void)out_size; (void)ws_size;

  char* ws = (char*)d_ws;
  size_t off = 0;
  auto alloc = [&](size_t bytes) -> float* {
    float* p = (float*)(ws + off);
    off += (bytes + 255) & ~(size_t)255;
    return p;
  };
  float* invn   = alloc((size_t)NNODES * 4);           //  0.2 MB
  float* bufA   = alloc((size_t)NNODES * HID * 4);     // 25.6 MB: X1 -> X2 -> Yemb
  float* bufAgg = alloc((size_t)NNODES * HID * 4);     // 25.6 MB: agg1 -> agg2
  float* bufL   = alloc((size_t)NNODES * NPAD * 4);    //  9.6 MB: logits
  float* Wtp    = alloc((size_t)HID * NPAD * 4);
  float* Wbp    = alloc((size_t)HID * NPAD * 4);
  float* bfp    = alloc((size_t)NPAD * 4);

  const int gemmBlocks = ((NNODES / 16) + 7) / 8;      // 391

  // 1) inverse row norms (fused normalize) + head weight prep
  k_rownorm<<<NNODES / 8, 256, 0, stream>>>(features, invn, NNODES, FIN);
  k_prep_head<<<1, 64, 0, stream>>>(Wf, bf, be, b2, Wtp, Wbp, bfp);

  // 2) X1 = normalize(features) @ W1          (row scale at store; MODE=1)
  k_gemm_wmma<8, FIN, 1><<<gemmBlocks, 256, 0, stream>>>(
      features, W1, bufA, NNODES, invn, nullptr, nullptr);
  // 3) agg1 = A @ X1
  k_zero<<<2048, 256, 0, stream>>>(bufAgg, (size_t)NNODES * HID / 4);
  k_spmm<<<NEDGES / 8, 256, 0, stream>>>(src, dst, ew, bufA, bufAgg, NEDGES);
  // 4) X2 = relu(agg1 + b1) @ W2              (bias+ReLU fused; MODE=2)
  k_gemm_wmma<8, HID, 2><<<gemmBlocks, 256, 0, stream>>>(
      bufAgg, W2, bufA, NNODES, nullptr, b1, nullptr);
  // 5) agg2 = A @ X2
  k_zero<<<2048, 256, 0, stream>>>(bufAgg, (size_t)NNODES * HID / 4);
  k_spmm<<<NEDGES / 8, 256, 0, stream>>>(src, dst, ew, bufA, bufAgg, NEDGES);
  // 6) Yemb = y @ We                          (be folded into bfp; MODE=0)
  k_gemm_wmma<8, NCLS, 0><<<gemmBlocks, 256, 0, stream>>>(
      y, We, bufA, NNODES, nullptr, nullptr, nullptr);
  // 7) logits = Yemb @ Wf_top + bfp (MODE=4) ; logits += agg2 @ Wf_bot (MODE=8)
  k_gemm_wmma<3, HID, 4><<<gemmBlocks, 256, 0, stream>>>(
      bufA, Wtp, bufL, NNODES, nullptr, nullptr, bfp);
  k_gemm_wmma<3, HID, 8><<<gemmBlocks, 256, 0, stream>>>(
      bufAgg, Wbp, bufL, NNODES, nullptr, nullptr, nullptr);
  // 8) softmax -> out
  k_softmax<<<NNODES / 8, 256, 0, stream>>>(bufL, out, NNODES);
}